// ConvSelfLayer_std_29240137351516
// MI455X (gfx1250) — compile-verified
//
#include <hip/hip_runtime.h>

// Self full-convolution: out[b,t] = sum_i x[b,i]*x[b,t-i], B=128, L=4096.
// Mapped onto V_WMMA_F32_16X16X4_F32 (full fp32 precision, matches reference).

#define BATCH 128
#define LEN   4096
#define OUTL  (2 * LEN - 1)        // 8191
#define PAD   1024
#define XBUF  (LEN + 2 * PAD)      // 6144 floats = 24 KB LDS

typedef float v2f __attribute__((ext_vector_type(2)));
typedef float v8f __attribute__((ext_vector_type(8)));

// Only check for the CDNA5 builtin in the device pass; the host pass
// (x86 clang) does not declare AMDGCN builtins.
#if defined(__HIP_DEVICE_COMPILE__)
#if !__has_builtin(__builtin_amdgcn_wmma_f32_16x16x4_f32)
#error "missing __builtin_amdgcn_wmma_f32_16x16x4_f32 on this toolchain"
#endif
#endif

__global__ __launch_bounds__(256) void selfconv_wmma_f32(
    const float* __restrict__ x, float* __restrict__ out) {
  __shared__ float X[XBUF];

  const int tid = threadIdx.x;
  const int b   = blockIdx.x;

  // --- Stage x[b,:] into LDS with zero halos (padding implements conv edges)
  for (int i = tid; i < 2 * PAD; i += 256) {
    int idx = (i < PAD) ? i : (LEN + i);   // [0,PAD) and [LEN+PAD, XBUF)
    X[idx] = 0.0f;
  }
  const float4* xb4 = (const float4*)(x + b * LEN);
  for (int i = tid; i < LEN / 4; i += 256) {
    float4 v = xb4[i];
    X[PAD + 4 * i + 0] = v.x;
    X[PAD + 4 * i + 1] = v.y;
    X[PAD + 4 * i + 2] = v.z;
    X[PAD + 4 * i + 3] = v.w;
  }
  __syncthreads();

  // --- Wave decomposition: wave w owns 4 output tiles tb = w*1024 + 256*j
  const int wave = tid >> 5;
  const int lane = tid & 31;
  const int m    = lane & 15;   // A-row M / B-col N / D-col N (per ISA layouts)
  const int kh   = lane >> 4;   // lane-half selects K pair {2kh, 2kh+1}
  const int gb   = wave << 10;  // group output base

  // Split-point range u: union over the 4 tiles, clipped to valid j-range.
  int u_lo = gb - (LEN - 1);
  if (u_lo < -240) u_lo = -240;
  u_lo &= ~3;                               // round down to K step
  int u_hi = gb + 783;
  if (u_hi > LEN - 1) u_hi = LEN - 1;

  // Per-lane fragment base offsets into padded LDS array.
  // A[m,k] = X[PAD + tb + m - u - k], k = 2*kh + {0,1}  (a.x, a.y)
  // B[k,n] = X[PAD + u + k + 16n],    k = 2*kh + {0,1}  (b.x, b.y)
  const int aBase = PAD + gb + m - 2 * kh;   // a.x index at u=0, tile 0
  const int bBase = PAD + 2 * kh + 16 * m;   // b.x index at u=0

  v8f acc[4];
  acc[0] = (v8f){0.f, 0.f, 0.f, 0.f, 0.f, 0.f, 0.f, 0.f};
  acc[1] = acc[0];
  acc[2] = acc[0];
  acc[3] = acc[0];

  for (int u = u_lo; u <= u_hi; u += 4) {
    v2f bf;
    bf.x = X[bBase + u];
    bf.y = X[bBase + u + 1];
#pragma unroll
    for (int j = 0; j < 4; ++j) {
      const int ai = aBase + 256 * j - u;
      v2f af;
      af.x = X[ai];
      af.y = X[ai - 1];
      // D = A(16x4,f32) x B(4x16,f32) + C(16x16,f32)
      acc[j] = __builtin_amdgcn_wmma_f32_16x16x4_f32(
          /*neg_a=*/false, af, /*neg_b=*/false, bf,
          /*c_mod=*/(short)0, acc[j], /*reuse_a=*/false, /*reuse_b=*/false);
    }
  }

  // --- Store: D lane layout -> t = tb + (r + 8*kh) + 16*m
  float* outb = out + b * OUTL;
#pragma unroll
  for (int j = 0; j < 4; ++j) {
    const int tb = gb + 256 * j;
#pragma unroll
    for (int r = 0; r < 8; ++r) {
      const int t = tb + r + 8 * kh + 16 * m;
      if (t < OUTL) outb[t] = acc[j][r];
    }
  }
}

extern "C" void kernel_launch(void* const* d_in, const int* in_sizes, int n_in,
                              void* d_out, int out_size, void* d_ws, size_t ws_size,
                              hipStream_t stream) {
  (void)in_sizes; (void)n_in; (void)d_ws; (void)ws_size; (void)out_size;
  const float* x = (const float*)d_in[0];
  float* out = (float*)d_out;
  selfconv_wmma_f32<<<dim3(BATCH), dim3(256), 0, stream>>>(x, out);
}